// WeightedMergeLayer_74096775791127
// MI455X (gfx1250) — compile-verified
//
#include <hip/hip_runtime.h>

typedef __attribute__((ext_vector_type(2))) float v2f;
typedef __attribute__((ext_vector_type(8))) float v8f;

#define D_IN  4096
#define D_H   4096
#define D_OUTP 1000
#define BATCH 64

__device__ __forceinline__ void softmax3(const float* __restrict__ A, int row,
                                         float& a0, float& a1, float& a2) {
  float v0 = A[row * 3 + 0], v1 = A[row * 3 + 1], v2 = A[row * 3 + 2];
  float mx = fmaxf(v0, fmaxf(v1, v2));
  float e0 = __expf(v0 - mx), e1 = __expf(v1 - mx), e2 = __expf(v2 - mx);
  float inv = 1.0f / (e0 + e1 + e2);
  a0 = e0 * inv; a1 = e1 * inv; a2 = e2 * inv;
}

// x[64,4096] (row major) -> xT[4096,64] so WMMA A tiles load contiguously.
__global__ void __launch_bounds__(256) transpose_x(const float* __restrict__ x,
                                                   float* __restrict__ xT) {
  int idx = blockIdx.x * 256 + threadIdx.x;   // 64*4096 total
  int b = idx >> 12;                          // / 4096
  int k = idx & 4095;
  xT[k * BATCH + b] = x[idx];
}

// H^T = relu(x @ (W1_base + sum_m a_m W1_tv[...,m]) + b1)^T, stored [D_H][64].
// One block per 16-column stripe of W1; 8 waves split K; LDS split-K reduce.
__global__ void __launch_bounds__(256) fused_gemm1(
    const float* __restrict__ xT, const float* __restrict__ alphaRaw,
    const float* __restrict__ W1b, const float* __restrict__ b1b,
    const float* __restrict__ W1tv, const float* __restrict__ b1tv,
    float* __restrict__ HT) {
  __shared__ float red[8][32][32];

  const int wave = threadIdx.x >> 5;
  const int lane = threadIdx.x & 31;
  const int ln   = lane & 15;
  const int koff = (lane >> 4) << 1;     // lanes 16-31 handle K+2,K+3
  const int h0   = blockIdx.x * 16;
  const int col  = h0 + ln;

  float a0, a1, a2;
  softmax3(alphaRaw, 0, a0, a1, a2);

  v8f z = 0.0f;
  v8f acc[4] = {z, z, z, z};

  const int kbeg = wave << 9;            // 512 K per wave
  for (int k = kbeg; k < kbeg + 512; k += 4) {
    const int kk = k + koff;
    const float* bb = W1b + (size_t)kk * D_H + col;
    const float* tt = W1tv + ((size_t)kk * D_H + col) * 3;
    __builtin_prefetch((const void*)(bb + 8 * D_H), 0, 0);
    __builtin_prefetch((const void*)(tt + 8 * D_H * 3), 0, 0);
    // merge two B elements (rows kk, kk+1 of column `col`)
    float mw0 = fmaf(a2, tt[2], fmaf(a1, tt[1], fmaf(a0, tt[0], bb[0])));
    const float* tt1 = tt + D_H * 3;
    float mw1 = fmaf(a2, tt1[2], fmaf(a1, tt1[1], fmaf(a0, tt1[0], bb[D_H])));
    v2f Bm; Bm.x = mw0; Bm.y = mw1;

    const float* xr0 = xT + kk * BATCH + ln;
    const float* xr1 = xr0 + BATCH;
#pragma unroll
    for (int mt = 0; mt < 4; ++mt) {
      v2f Am; Am.x = xr0[mt * 16]; Am.y = xr1[mt * 16];
      acc[mt] = __builtin_amdgcn_wmma_f32_16x16x4_f32(
          false, Am, false, Bm, (short)0, acc[mt], false, false);
    }
  }

  // stash per-wave partial C (64x16) into LDS: red[wave][mt*8+r][lane]
#pragma unroll
  for (int mt = 0; mt < 4; ++mt)
#pragma unroll
    for (int r = 0; r < 8; ++r)
      red[wave][mt * 8 + r][lane] = acc[mt][r];
  __syncthreads();

  // reduce 8 partials, add merged bias, relu, store transposed (coalesced on b)
  const int b  = threadIdx.x & 63;
  const int c0 = threadIdx.x >> 6;       // 0..3
  const int lane_e = ((b & 15) >= 8) ? 16 : 0;
  const int i_e = ((b >> 4) << 3) + (b & 7);
#pragma unroll
  for (int j = 0; j < 4; ++j) {
    int c = c0 + j * 4;
    float s = 0.0f;
#pragma unroll
    for (int w = 0; w < 8; ++w) s += red[w][i_e][c + lane_e];
    int h = h0 + c;
    float b0v, b1v, b2v;
    softmax3(alphaRaw, 1, b0v, b1v, b2v);
    float bias = b1b[h] + b0v * b1tv[h * 3] + b1v * b1tv[h * 3 + 1] + b2v * b1tv[h * 3 + 2];
    s = fmaxf(s + bias, 0.0f);
    HT[(size_t)h * BATCH + b] = s;
  }
}

// out = beta * (H @ (W2_base + sum_m a_m W2_tv[...,m]) + b2), out[64,1000]
__global__ void __launch_bounds__(256) fused_gemm2(
    const float* __restrict__ HT, const float* __restrict__ alphaRaw,
    const float* __restrict__ betap,
    const float* __restrict__ W2b, const float* __restrict__ b2b,
    const float* __restrict__ W2tv, const float* __restrict__ b2tv,
    float* __restrict__ out) {
  __shared__ float red[8][32][32];

  const int wave = threadIdx.x >> 5;
  const int lane = threadIdx.x & 31;
  const int ln   = lane & 15;
  const int koff = (lane >> 4) << 1;
  const int h0   = blockIdx.x * 16;                 // 63 blocks cover 1000 (+8 pad)
  const int col  = min(h0 + ln, D_OUTP - 1);        // clamp loads (stores guarded)

  float a0, a1, a2;
  softmax3(alphaRaw, 2, a0, a1, a2);

  v8f z = 0.0f;
  v8f acc[4] = {z, z, z, z};

  const int kbeg = wave << 9;
  for (int k = kbeg; k < kbeg + 512; k += 4) {
    const int kk = k + koff;
    const float* bb = W2b + (size_t)kk * D_OUTP + col;
    const float* tt = W2tv + ((size_t)kk * D_OUTP + col) * 3;
    __builtin_prefetch((const void*)(bb + 8 * D_OUTP), 0, 0);
    __builtin_prefetch((const void*)(tt + 8 * D_OUTP * 3), 0, 0);
    float mw0 = fmaf(a2, tt[2], fmaf(a1, tt[1], fmaf(a0, tt[0], bb[0])));
    const float* tt1 = tt + D_OUTP * 3;
    float mw1 = fmaf(a2, tt1[2], fmaf(a1, tt1[1], fmaf(a0, tt1[0], bb[D_OUTP])));
    v2f Bm; Bm.x = mw0; Bm.y = mw1;

    const float* hr0 = HT + kk * BATCH + ln;
    const float* hr1 = hr0 + BATCH;
#pragma unroll
    for (int mt = 0; mt < 4; ++mt) {
      v2f Am; Am.x = hr0[mt * 16]; Am.y = hr1[mt * 16];
      acc[mt] = __builtin_amdgcn_wmma_f32_16x16x4_f32(
          false, Am, false, Bm, (short)0, acc[mt], false, false);
    }
  }

#pragma unroll
  for (int mt = 0; mt < 4; ++mt)
#pragma unroll
    for (int r = 0; r < 8; ++r)
      red[wave][mt * 8 + r][lane] = acc[mt][r];
  __syncthreads();

  // reduce + bias + beta; map threads so output columns are lane-contiguous
  const float beta = betap[0];
  const int c  = threadIdx.x & 15;
  const int b0 = threadIdx.x >> 4;       // 0..15
#pragma unroll
  for (int j = 0; j < 4; ++j) {
    int b = b0 + j * 16;
    int lane_e = c + (((b & 15) >= 8) ? 16 : 0);
    int i_e = ((b >> 4) << 3) + (b & 7);
    float s = 0.0f;
#pragma unroll
    for (int w = 0; w < 8; ++w) s += red[w][i_e][lane_e];
    int o = h0 + c;
    if (o < D_OUTP) {
      float b0v, b1v, b2v;
      softmax3(alphaRaw, 3, b0v, b1v, b2v);
      float bias = b2b[o] + b0v * b2tv[o * 3] + b1v * b2tv[o * 3 + 1] + b2v * b2tv[o * 3 + 2];
      out[(size_t)b * D_OUTP + o] = beta * (s + bias);
    }
  }
}

extern "C" void kernel_launch(void* const* d_in, const int* in_sizes, int n_in,
                              void* d_out, int out_size, void* d_ws, size_t ws_size,
                              hipStream_t stream) {
  const float* x      = (const float*)d_in[0];
  const float* alphaR = (const float*)d_in[1];
  const float* beta   = (const float*)d_in[2];
  const float* W1b    = (const float*)d_in[3];
  const float* b1b    = (const float*)d_in[4];
  const float* W2b    = (const float*)d_in[5];
  const float* b2b    = (const float*)d_in[6];
  const float* W1tv   = (const float*)d_in[7];
  const float* b1tv   = (const float*)d_in[8];
  const float* W2tv   = (const float*)d_in[9];
  const float* b2tv   = (const float*)d_in[10];
  float* out = (float*)d_out;

  float* xT = (float*)d_ws;                 // [4096][64] = 1 MB
  float* HT = xT + (size_t)D_IN * BATCH;    // [4096][64] = 1 MB

  transpose_x<<<(BATCH * D_IN) / 256, 256, 0, stream>>>(x, xT);
  fused_gemm1<<<D_H / 16, 256, 0, stream>>>(xT, alphaR, W1b, b1b, W1tv, b1tv, HT);
  fused_gemm2<<<(D_OUTP + 15) / 16, 256, 0, stream>>>(HT, alphaR, beta, W2b, b2b, W2tv, b2tv, out);
}